// GATModel_30434138259920
// MI455X (gfx1250) — compile-verified
//
#include <hip/hip_runtime.h>
#include <hip/hip_bf16.h>

#define N_NODES 50000
#define N_EDGES 800000
#define HID 128
#define OUT_DIM 16
#define NEG_SLOPE 0.2f

typedef __attribute__((ext_vector_type(2))) float v2f;
typedef __attribute__((ext_vector_type(8))) float v8f;

// ---------------------------------------------------------------------------
// Node embedding: h[n,d] = relu(key_emb[f0,d] + val_emb[f1,d])
// ---------------------------------------------------------------------------
__global__ __launch_bounds__(HID) void embed_kernel(
    const int* __restrict__ feats, const float* __restrict__ key_emb,
    const float* __restrict__ val_emb, float* __restrict__ h) {
  int n = blockIdx.x;
  int d = threadIdx.x;
  int f0 = feats[n * 2 + 0];
  int f1 = feats[n * 2 + 1];
  float v = key_emb[f0 * HID + d] + val_emb[f1 * HID + d];
  h[n * HID + d] = v > 0.f ? v : 0.f;
}

// ---------------------------------------------------------------------------
// feat = h @ W  (N x 128 @ 128 x 128), fp32 WMMA 16x16x4.
// Block = 256 threads = 8 waves; block covers one 16-row stripe, wave w does
// columns [16w, 16w+16). Grid = N/16 = 3125 blocks (exact).
// ---------------------------------------------------------------------------
__global__ __launch_bounds__(256) void gemm128_kernel(
    const float* __restrict__ A, const float* __restrict__ B,
    float* __restrict__ C) {
  const int wave = threadIdx.x >> 5;
  const int lane = threadIdx.x & 31;
  const int m0 = blockIdx.x * 16;
  const int n0 = wave * 16;
  const int r = lane & 15;          // row within tile (A) / col within tile (B)
  const int hi = lane >> 4;         // 0: lanes 0-15, 1: lanes 16-31
  const int koff = hi << 1;         // K sub-offset 0 or 2

  const float* arow = A + (size_t)(m0 + r) * HID;
  v8f acc = {};
#pragma unroll 8
  for (int k0 = 0; k0 < HID; k0 += 4) {
    const int k = k0 + koff;
    v2f a;
    a.x = arow[k];
    a.y = arow[k + 1];
    v2f b;
    b.x = B[k * HID + n0 + r];
    b.y = B[(k + 1) * HID + n0 + r];
    acc = __builtin_amdgcn_wmma_f32_16x16x4_f32(false, a, false, b, (short)0,
                                                acc, false, false);
  }
#pragma unroll
  for (int v = 0; v < 8; ++v) {
    C[(size_t)(m0 + v + (hi << 3)) * HID + n0 + r] = acc[v];
  }
}

// ---------------------------------------------------------------------------
// out = h @ cls_w (N x 128 @ 128 x 16). One wave per 16-row tile.
// ---------------------------------------------------------------------------
__global__ __launch_bounds__(256) void gemm_cls_kernel(
    const float* __restrict__ A, const float* __restrict__ B,
    float* __restrict__ C) {
  const int wave = threadIdx.x >> 5;
  const int tile = blockIdx.x * 8 + wave;   // uniform per wave
  if (tile >= N_NODES / 16) return;         // wave-uniform exit, EXEC stays full
  const int lane = threadIdx.x & 31;
  const int m0 = tile * 16;
  const int r = lane & 15;
  const int hi = lane >> 4;
  const int koff = hi << 1;

  const float* arow = A + (size_t)(m0 + r) * HID;
  v8f acc = {};
#pragma unroll 8
  for (int k0 = 0; k0 < HID; k0 += 4) {
    const int k = k0 + koff;
    v2f a;
    a.x = arow[k];
    a.y = arow[k + 1];
    v2f b;
    b.x = B[k * OUT_DIM + r];
    b.y = B[(k + 1) * OUT_DIM + r];
    acc = __builtin_amdgcn_wmma_f32_16x16x4_f32(false, a, false, b, (short)0,
                                                acc, false, false);
  }
#pragma unroll
  for (int v = 0; v < 8; ++v) {
    C[(size_t)(m0 + v + (hi << 3)) * OUT_DIM + r] = acc[v];
  }
}

// ---------------------------------------------------------------------------
// Per-node init: segment-max identity (-inf as ordered int) and denom = 0.
// ---------------------------------------------------------------------------
__device__ __forceinline__ int float_to_ord(float f) {
  int i = __float_as_int(f);
  return i >= 0 ? i : (i ^ 0x7FFFFFFF);
}
__device__ __forceinline__ float ord_to_float(int i) {
  return __int_as_float(i >= 0 ? i : (i ^ 0x7FFFFFFF));
}

__global__ void init_nodes_kernel(int* __restrict__ mmax,
                                  float* __restrict__ denom) {
  int i = blockIdx.x * 256 + threadIdx.x;
  if (i < N_NODES) {
    mmax[i] = float_to_ord(-__builtin_inff());
    denom[i] = 0.f;
  }
}

__global__ void zero_kernel(float* __restrict__ p, int n) {
  int i = blockIdx.x * 256 + threadIdx.x;
  if (i < n) p[i] = 0.f;
}

// ---------------------------------------------------------------------------
// Pass 1: logits[e] = attn . leaky_relu(feat[src] + feat[dst]); seg-max(dst).
// One wave per edge; lane handles a float4 chunk of the 128-dim feature.
// ---------------------------------------------------------------------------
__global__ __launch_bounds__(256) void edge_logits_kernel(
    const float* __restrict__ feat, const float* __restrict__ attn,
    const int* __restrict__ src, const int* __restrict__ dst,
    float* __restrict__ logits, int* __restrict__ mmax) {
  const int e = blockIdx.x * 8 + (threadIdx.x >> 5);
  const int lane = threadIdx.x & 31;
  const int s = src[e];
  const int d = dst[e];
  const float4 x = ((const float4*)(feat + (size_t)s * HID))[lane];
  const float4 y = ((const float4*)(feat + (size_t)d * HID))[lane];
  const float4 a4 = ((const float4*)attn)[lane];
  float p = 0.f, t;
  t = x.x + y.x; p += a4.x * (t > 0.f ? t : NEG_SLOPE * t);
  t = x.y + y.y; p += a4.y * (t > 0.f ? t : NEG_SLOPE * t);
  t = x.z + y.z; p += a4.z * (t > 0.f ? t : NEG_SLOPE * t);
  t = x.w + y.w; p += a4.w * (t > 0.f ? t : NEG_SLOPE * t);
#pragma unroll
  for (int off = 16; off > 0; off >>= 1) p += __shfl_xor(p, off, 32);
  if (lane == 0) {
    logits[e] = p;
    atomicMax(mmax + d, float_to_ord(p));
  }
}

// ---------------------------------------------------------------------------
// Pass 2: ex = exp(logit - max[dst]); denom[dst] += ex.  (ex overwrites logits)
// ---------------------------------------------------------------------------
__global__ __launch_bounds__(256) void edge_exp_kernel(
    const int* __restrict__ dst, float* __restrict__ logits,
    const int* __restrict__ mmax, float* __restrict__ denom) {
  int e = blockIdx.x * 256 + threadIdx.x;
  if (e >= N_EDGES) return;
  int d = dst[e];
  float mx = ord_to_float(mmax[d]);
  float ex = expf(logits[e] - mx);
  logits[e] = ex;
  atomicAdd(denom + d, ex);
}

// ---------------------------------------------------------------------------
// Pass 3: out[dst] += (ex/denom[dst]) * feat[src].  Wave per edge, float
// atomics to L2 (global_atomic_add_f32).
// ---------------------------------------------------------------------------
__global__ __launch_bounds__(256) void edge_aggr_kernel(
    const float* __restrict__ feat, const int* __restrict__ src,
    const int* __restrict__ dst, const float* __restrict__ ex,
    const float* __restrict__ denom, float* __restrict__ out) {
  const int e = blockIdx.x * 8 + (threadIdx.x >> 5);
  const int lane = threadIdx.x & 31;
  const int s = src[e];
  const int d = dst[e];
  const float a = ex[e] / denom[d];
  const float4 x = ((const float4*)(feat + (size_t)s * HID))[lane];
  float* o = out + (size_t)d * HID + lane * 4;
  atomicAdd(o + 0, a * x.x);
  atomicAdd(o + 1, a * x.y);
  atomicAdd(o + 2, a * x.z);
  atomicAdd(o + 3, a * x.w);
}

// ---------------------------------------------------------------------------
// Launcher
// ---------------------------------------------------------------------------
static void run_gat_layer(const float* h_in, const float* W, const float* attn,
                          const int* src, const int* dst, float* feat,
                          float* h_out, float* ex, int* mmax, float* denom,
                          hipStream_t stream) {
  gemm128_kernel<<<N_NODES / 16, 256, 0, stream>>>(h_in, W, feat);
  init_nodes_kernel<<<(N_NODES + 255) / 256, 256, 0, stream>>>(mmax, denom);
  zero_kernel<<<(N_NODES * HID + 255) / 256, 256, 0, stream>>>(h_out,
                                                               N_NODES * HID);
  edge_logits_kernel<<<N_EDGES / 8, 256, 0, stream>>>(feat, attn, src, dst, ex,
                                                      mmax);
  edge_exp_kernel<<<(N_EDGES + 255) / 256, 256, 0, stream>>>(dst, ex, mmax,
                                                             denom);
  edge_aggr_kernel<<<N_EDGES / 8, 256, 0, stream>>>(feat, src, dst, ex, denom,
                                                    h_out);
}

extern "C" void kernel_launch(void* const* d_in, const int* in_sizes, int n_in,
                              void* d_out, int out_size, void* d_ws,
                              size_t ws_size, hipStream_t stream) {
  const int* feats = (const int*)d_in[0];     // [N,2] int32
  const int* src = (const int*)d_in[1];       // [E]
  const int* dst = (const int*)d_in[2];       // [E]
  const float* key_emb = (const float*)d_in[3];
  const float* val_emb = (const float*)d_in[4];
  const float* W1 = (const float*)d_in[5];
  const float* attn1 = (const float*)d_in[6];
  const float* W2 = (const float*)d_in[7];
  const float* attn2 = (const float*)d_in[8];
  const float* cls_w = (const float*)d_in[9];
  float* out = (float*)d_out;                 // [N,16]

  // Workspace carve-up (~80.4 MB total)
  float* hA = (float*)d_ws;                   // N*HID
  float* feat = hA + (size_t)N_NODES * HID;   // N*HID
  float* hB = feat + (size_t)N_NODES * HID;   // N*HID
  float* ex = hB + (size_t)N_NODES * HID;     // E (logits then exp)
  int* mmax = (int*)(ex + N_EDGES);           // N
  float* denom = (float*)(mmax + N_NODES);    // N

  // h = relu(key_emb[f0] + val_emb[f1])
  embed_kernel<<<N_NODES, HID, 0, stream>>>(feats, key_emb, val_emb, hA);

  // Layer 1: hA -> hB
  run_gat_layer(hA, W1, attn1, src, dst, feat, hB, ex, mmax, denom, stream);
  // Layer 2: hB -> hA
  run_gat_layer(hB, W2, attn2, src, dst, feat, hA, ex, mmax, denom, stream);

  // Classifier: out = hA @ cls_w
  gemm_cls_kernel<<<(N_NODES / 16 + 7) / 8, 256, 0, stream>>>(hA, cls_w, out);
}